// FMLayer_37014028157489
// MI455X (gfx1250) — compile-verified
//
#include <hip/hip_runtime.h>

typedef __attribute__((ext_vector_type(2))) float    v2f;
typedef __attribute__((ext_vector_type(4))) float    v4f;
typedef __attribute__((ext_vector_type(8))) float    v8f;
typedef __attribute__((ext_vector_type(4))) unsigned v4u;

#define NF      512
#define KLAT    4
#define BATCH   1024
#define NPAIRS  130816          /* 512*511/2 */
#define NP4     (NPAIRS / 4)    /* 32704, exact */
#define BPB     8               /* batch rows per block in kernel 2 */

// ---------------------------------------------------------------------------
// Kernel 1: G = W * W^T via v_wmma_f32_16x16x4_f32 (one wave = one 16x16 tile),
// scattering the strict upper triangle straight into flat pair tables:
//   wdot[p] = <W[i1,:], W[i2,:]>      idx[p] = (i1<<16) | i2
// ---------------------------------------------------------------------------
__global__ __launch_bounds__(32)
void fm_wdot_wmma(const float* __restrict__ W,
                  float* __restrict__ wdot,
                  unsigned* __restrict__ idx) {
  const int ti = blockIdx.y;          // row tile  (i1 / 16)
  const int tj = blockIdx.x;          // col tile  (i2 / 16)
  if (tj < ti) return;                // whole-wave exit: EXEC stays all-1s

  const int lane = threadIdx.x;       // wave32
  const int half = lane >> 4;         // 0: K={0,1}   1: K={2,3}
  const int l16  = lane & 15;
  const int kb   = half * 2;

  // A tile: rows (ti*16 + l16) of W, 2 consecutive K per lane (contiguous 8B)
  const int rA = ti * 16 + l16;
  const int rB = tj * 16 + l16;       // B = W^T  ->  same per-lane load shape
  v2f a, b;
  a.x = W[rA * KLAT + kb];  a.y = W[rA * KLAT + kb + 1];
  b.x = W[rB * KLAT + kb];  b.y = W[rB * KLAT + kb + 1];

  v8f c = {};
  c = __builtin_amdgcn_wmma_f32_16x16x4_f32(
        /*neg_a=*/false, a, /*neg_b=*/false, b,
        /*c_mod=*/(short)0, c, /*reuse_a=*/false, /*reuse_b=*/false);

  // C/D layout: VGPR r, lane L  ->  (M = r + 8*(L/16), N = L%16)
  const int gj     = tj * 16 + l16;
  const int giBase = ti * 16 + half * 8;
#pragma unroll
  for (int r = 0; r < 8; ++r) {
    const int gi = giBase + r;
    if (gi < gj) {
      // triu row offset: off(gi) = gi*(1023-gi)/2  (product always even)
      const int p = (gi * (2 * NF - 1 - gi)) / 2 + (gj - gi - 1);
      wdot[p] = c[r];
      idx[p]  = ((unsigned)gi << 16) | (unsigned)gj;
    }
  }
}

// ---------------------------------------------------------------------------
// Kernel 2: the 536 MB streaming kernel (store-bandwidth bound, ~23 us floor).
// grid = (128 pair-chunks, 128 batch-groups). Each block stages 8 x-rows
// (16 KB LDS), loads its idx/wdot vectors ONCE into registers, then emits
// 8 x 128-bit non-temporal stores — table reads amortized 8x, dense NT store
// issue, output stream bypasses far cache so the 1 MB tables stay L2-resident.
// ---------------------------------------------------------------------------
__global__ __launch_bounds__(256)
void fm_out(const float* __restrict__ x,
            const float* __restrict__ wdot,
            const unsigned* __restrict__ idx,
            float* __restrict__ out) {
  __shared__ float sx[BPB][NF];
  const int t  = threadIdx.x;         // 0..255
  const int b0 = blockIdx.y * BPB;    // first batch row of this block

  // coalesced LDS stage: 8 rows * 512 floats = 4096 floats = 4 x v4f / thread
  const v4f* xsrc = reinterpret_cast<const v4f*>(x + (size_t)b0 * NF);
  v4f*       sdst = reinterpret_cast<v4f*>(&sx[0][0]);
#pragma unroll
  for (int i = 0; i < (BPB * NF / 4) / 256; ++i)   // 4 iterations
    sdst[t + i * 256] = xsrc[t + i * 256];
  __syncthreads();

  const int p4 = blockIdx.x * 256 + t;
  if (p4 >= NP4) return;              // tail guard (last pair-chunk: 192 active)
  const int p = p4 * 4;

  const v4u id = *reinterpret_cast<const v4u*>(idx + p);   // once per 8 rows
  const v4f wd = *reinterpret_cast<const v4f*>(wdot + p);  // once per 8 rows
  const int i1x = id.x >> 16, i2x = id.x & 0xFFFF;
  const int i1y = id.y >> 16, i2y = id.y & 0xFFFF;
  const int i1z = id.z >> 16, i2z = id.z & 0xFFFF;
  const int i1w = id.w >> 16, i2w = id.w & 0xFFFF;

#pragma unroll
  for (int bb = 0; bb < BPB; ++bb) {
    const float* s = sx[bb];
    v4f o;
    o.x = s[i1x] * s[i2x] * wd.x;
    o.y = s[i1y] * s[i2y] * wd.y;
    o.z = s[i1z] * s[i2z] * wd.z;
    o.w = s[i1w] * s[i2w] * wd.w;
    v4f* dst = reinterpret_cast<v4f*>(out + (size_t)(b0 + bb) * NPAIRS + p);
    __builtin_nontemporal_store(o, dst);   // NT: don't dirty far cache
  }
}

// ---------------------------------------------------------------------------
extern "C" void kernel_launch(void* const* d_in, const int* in_sizes, int n_in,
                              void* d_out, int out_size, void* d_ws, size_t ws_size,
                              hipStream_t stream) {
  const float* x = (const float*)d_in[0];   // [1024, 512] f32
  const float* w = (const float*)d_in[1];   // [512, 4]   f32
  float* out = (float*)d_out;               // [1024, 130816] f32

  // workspace: wdot (P floats) then idx (P u32)  -> ~1 MB, 16B-aligned slices
  float*    wdot = (float*)d_ws;
  unsigned* idx  = (unsigned*)((char*)d_ws + (size_t)NPAIRS * sizeof(float));

  dim3 g1(NF / 16, NF / 16), b1(32);        // 32x32 tiles, 1 wave each
  fm_wdot_wmma<<<g1, b1, 0, stream>>>(w, wdot, idx);

  dim3 g2((NP4 + 255) / 256, BATCH / BPB), b2(256);
  fm_out<<<g2, b2, 0, stream>>>(x, wdot, idx, out);
}